// MixtureOfExperts_28020366639543
// MI455X (gfx1250) — compile-verified
//
#include <hip/hip_runtime.h>
#include <hip/hip_bf16.h>

// Problem dims (from reference): B=4, S=2048, D=1024, F=4096, E=8, K=2
#define T_TOK 8192
#define DDIM  1024
#define FDIM  4096
#define NEXP  8

// FFN tiling
#define MT   32          // tokens per workgroup (2 x 16-row WMMA tiles)
#define FC   128         // F-chunk staged in LDS per iteration
#define XST  1028        // LDS row stride for x tile (1024 + 4 pad -> stride%64==4)
#define HST  132         // LDS row stride for h tile (128 + 4 pad)

typedef __attribute__((ext_vector_type(2))) float v2f;
typedef __attribute__((ext_vector_type(8))) float v8f;

// D = A(16x4,f32) * B(4x16,f32) + C(16x16,f32), wave32 WMMA
__device__ __forceinline__ v8f wmma4(v2f a, v2f b, v8f c) {
    return __builtin_amdgcn_wmma_f32_16x16x4_f32(
        /*neg_a=*/false, a, /*neg_b=*/false, b,
        /*c_mod=*/(short)0, c, /*reuse_a=*/false, /*reuse_b=*/false);
}

__device__ __forceinline__ float gelu_exact(float v) {
    // exact GELU: x * 0.5 * (1 + erf(x / sqrt(2)))
    return 0.5f * v * (1.0f + erff(v * 0.70710678118654752f));
}

// ---------------------------------------------------------------------------
// Kernel 0: zero output accumulator + per-expert counters
// ---------------------------------------------------------------------------
__global__ void moe_zero(float4* __restrict__ out4, int* __restrict__ cnt, int n4) {
    int i = blockIdx.x * blockDim.x + threadIdx.x;
    if (i < n4) out4[i] = make_float4(0.f, 0.f, 0.f, 0.f);
    if (i < NEXP) cnt[i] = 0;
}

// ---------------------------------------------------------------------------
// Kernel 1: gating. One wave (32 lanes) per token.
//   logits = x @ Wg + bg ; top-2 ; softmax over the 2 ; push to expert lists.
// ---------------------------------------------------------------------------
__global__ __launch_bounds__(256) void moe_gate(const float* __restrict__ x,
                                                const float* __restrict__ Wg,
                                                const float* __restrict__ bg,
                                                int* __restrict__ cnt,
                                                int* __restrict__ list,
                                                float* __restrict__ slist) {
    const int wave = threadIdx.x >> 5;
    const int lane = threadIdx.x & 31;
    const int t = blockIdx.x * 8 + wave;

    float a[NEXP];
#pragma unroll
    for (int e = 0; e < NEXP; ++e) a[e] = 0.0f;

    const float* xr = x + (size_t)t * DDIM;
    for (int d = lane; d < DDIM; d += 32) {
        const float xv = xr[d];
        const float* wr = Wg + (size_t)d * NEXP;
#pragma unroll
        for (int e = 0; e < NEXP; ++e) a[e] = fmaf(xv, wr[e], a[e]);
    }
    // cross-lane reduction (wave32)
#pragma unroll
    for (int e = 0; e < NEXP; ++e) {
#pragma unroll
        for (int off = 16; off > 0; off >>= 1) a[e] += __shfl_xor(a[e], off, 32);
    }

    if (lane == 0) {
#pragma unroll
        for (int e = 0; e < NEXP; ++e) a[e] += bg[e];
        // top-1 (first index wins ties, matching jax.lax.top_k)
        int i0 = 0; float v0 = a[0];
#pragma unroll
        for (int e = 1; e < NEXP; ++e) { if (a[e] > v0) { v0 = a[e]; i0 = e; } }
        // top-2
        int i1 = -1; float v1 = -3.4e38f;
#pragma unroll
        for (int e = 0; e < NEXP; ++e) {
            if (e != i0 && a[e] > v1) { v1 = a[e]; i1 = e; }
        }
        // softmax over {v0, v1} (shift by v0)
        const float e1 = __expf(v1 - v0);
        const float inv = 1.0f / (1.0f + e1);
        const float s0 = inv, s1 = e1 * inv;

        int p0 = atomicAdd(&cnt[i0], 1);
        list[i0 * T_TOK + p0]  = t;
        slist[i0 * T_TOK + p0] = s0;
        int p1 = atomicAdd(&cnt[i1], 1);
        list[i1 * T_TOK + p1]  = t;
        slist[i1 * T_TOK + p1] = s1;
    }
}

// ---------------------------------------------------------------------------
// Kernel 2: routed expert FFN with f32 WMMA.
//   grid.x = token tiles (worst case T/MT), grid.y = expert
//   Per WG: 32 tokens (2 M-tiles), full D and F of one expert.
//   8 waves: wave w owns column group w and processes BOTH M-tiles, so every
//   B fragment (weights) is reused for 2 WMMAs -> halves L2 weight traffic.
//     GEMM1: f-cols [w*16, w*16+16) of the current F-chunk (1 N-tile x 2 M)
//     GEMM2: d-cols [w*128, w*128+128)  (8 N-tiles x 2 M, acc in VGPRs)
// ---------------------------------------------------------------------------
__global__ __launch_bounds__(256) void moe_ffn(const float* __restrict__ x,
                                               const float* __restrict__ W1,
                                               const float* __restrict__ b1,
                                               const float* __restrict__ W2,
                                               const float* __restrict__ b2,
                                               const int* __restrict__ cnt,
                                               const int* __restrict__ list,
                                               const float* __restrict__ slist,
                                               float* __restrict__ out) {
    const int e    = blockIdx.y;
    const int tile = blockIdx.x;
    const int count = cnt[e];
    const int t0 = tile * MT;
    if (t0 >= count) return;
    const int nTok = min(MT, count - t0);

    extern __shared__ float smem[];
    float* xs = smem;                       // [MT][XST]
    float* hs = xs + MT * XST;              // [MT][HST]
    float* sc = hs + MT * HST;              // [MT] gate scores
    int*   tk = (int*)(sc + MT);            // [MT] token indices

    const int tid  = threadIdx.x;
    const int lane = tid & 31;
    const int wg   = tid >> 5;              // wave id == column group 0..7
    const int lhi  = lane >> 4;             // 0/1: lane half (K split)
    const int lm   = lane & 15;             // 0..15

    if (tid < MT) {
        const int p  = t0 + tid;
        const int pc = (p < count) ? p : (count - 1);
        tk[tid] = list[e * T_TOK + pc];
        sc[tid] = (p < count) ? slist[e * T_TOK + pc] : 0.0f;
    }
    __syncthreads();

    // Stage the 32 x rows into LDS (coalesced float4 per row).
#pragma unroll 4
    for (int row = 0; row < MT; ++row) {
        const float4 v = *(const float4*)(x + (size_t)tk[row] * DDIM + tid * 4);
        *(float4*)(xs + row * XST + tid * 4) = v;
    }
    __syncthreads();

    const size_t W1e = (size_t)e * DDIM * FDIM;
    const size_t W2e = (size_t)e * FDIM * DDIM;

    v8f acc[2][8];                          // [m-tile][n-tile]
#pragma unroll
    for (int m = 0; m < 2; ++m)
#pragma unroll
        for (int nt = 0; nt < 8; ++nt) acc[m][nt] = (v8f)0.0f;

    // A-fragment base pointers for the two M tiles (shared B fragments)
    const float* xrow0 = xs + (lm)      * XST + 2 * lhi;
    const float* xrow1 = xs + (16 + lm) * XST + 2 * lhi;

    for (int fc0 = 0; fc0 < FDIM; fc0 += FC) {
        // ---------------- GEMM1: h = x @ W1[:, fc0 + wg*16 .. +16) ----------
        v8f h0 = (v8f)0.0f, h1 = (v8f)0.0f;
        const float* w1p = W1 + W1e + (size_t)(2 * lhi) * FDIM
                              + (size_t)fc0 + wg * 16 + lm;
        for (int kb = 0; kb < DDIM / 4; ++kb) {
            const v2f a0 = *(const v2f*)(xrow0 + kb * 4);        // ds_load_b64
            const v2f a1 = *(const v2f*)(xrow1 + kb * 4);
            const float* bp = w1p + (size_t)(kb * 4) * FDIM;
            v2f bv;
            bv[0] = bp[0];
            bv[1] = bp[FDIM];
            h0 = wmma4(a0, bv, h0);                              // B reused x2
            h1 = wmma4(a1, bv, h1);
        }
        // bias + exact GELU -> hs
        {
            const int   fb = fc0 + wg * 16 + lm;
            const float bb = b1[(size_t)e * FDIM + fb];
            const int mb0 = 8 * lhi;
            const int mb1 = 16 + 8 * lhi;
#pragma unroll
            for (int r = 0; r < 8; ++r) {
                hs[(mb0 + r) * HST + wg * 16 + lm] = gelu_exact(h0[r] + bb);
                hs[(mb1 + r) * HST + wg * 16 + lm] = gelu_exact(h1[r] + bb);
            }
        }
        __syncthreads();   // hs ready for everyone

        // ---------------- GEMM2: out += h_chunk @ W2[fc0.., d-cols] ---------
        const float* hrow0 = hs + (lm)      * HST + 2 * lhi;
        const float* hrow1 = hs + (16 + lm) * HST + 2 * lhi;
        const float* w2p   = W2 + W2e + (size_t)(fc0 + 2 * lhi) * DDIM
                                + wg * 128 + lm;
        for (int kb = 0; kb < FC / 4; ++kb) {
            const v2f a0 = *(const v2f*)(hrow0 + kb * 4);
            const v2f a1 = *(const v2f*)(hrow1 + kb * 4);
            const float* bp = w2p + (size_t)(kb * 4) * DDIM;
#pragma unroll
            for (int nt = 0; nt < 8; ++nt) {
                v2f bv;
                bv[0] = bp[nt * 16];
                bv[1] = bp[DDIM + nt * 16];
                acc[0][nt] = wmma4(a0, bv, acc[0][nt]);          // B reused x2
                acc[1][nt] = wmma4(a1, bv, acc[1][nt]);
            }
        }
        __syncthreads();   // hs consumed; next chunk may overwrite
    }

    // Epilogue: out[t, d] += score_t * (acc + b2[e, d]); exactly K=2 adds per
    // element over the whole grid -> commutative, deterministic.
#pragma unroll
    for (int nt = 0; nt < 8; ++nt) {
        const int   d  = wg * 128 + nt * 16 + lm;
        const float bb = b2[(size_t)e * DDIM + d];
#pragma unroll
        for (int m = 0; m < 2; ++m) {
            const int mbase = m * 16 + 8 * lhi;
#pragma unroll
            for (int r = 0; r < 8; ++r) {
                const int mrow = mbase + r;
                if (mrow < nTok) {
                    const float v = (acc[m][nt][r] + bb) * sc[mrow];
                    atomicAdd(out + (size_t)tk[mrow] * DDIM + d, v);
                }
            }
        }
    }
}

// ---------------------------------------------------------------------------
extern "C" void kernel_launch(void* const* d_in, const int* in_sizes, int n_in,
                              void* d_out, int out_size, void* d_ws, size_t ws_size,
                              hipStream_t stream) {
    const float* x  = (const float*)d_in[0];
    const float* Wg = (const float*)d_in[1];
    const float* bg = (const float*)d_in[2];
    const float* W1 = (const float*)d_in[3];
    const float* b1 = (const float*)d_in[4];
    const float* W2 = (const float*)d_in[5];
    const float* b2 = (const float*)d_in[6];
    float* out = (float*)d_out;

    // Workspace layout: [cnt: 16 ints][list: E*T ints][slist: E*T floats]
    int*   cnt   = (int*)d_ws;
    int*   list  = cnt + 16;
    float* slist = (float*)(list + NEXP * T_TOK);

    const int n4 = (T_TOK * DDIM) / 4;
    moe_zero<<<(n4 + 255) / 256, 256, 0, stream>>>((float4*)out, cnt, n4);

    moe_gate<<<T_TOK / 8, 256, 0, stream>>>(x, Wg, bg, cnt, list, slist);

    const size_t lds = (size_t)(MT * XST + MT * HST + MT) * sizeof(float)
                     + (size_t)MT * sizeof(int);
    dim3 grid(T_TOK / MT, NEXP);
    moe_ffn<<<grid, 256, lds, stream>>>(x, W1, b1, W2, b2, cnt, list, slist, out);
}